// BaseRenderer_62569083568726
// MI455X (gfx1250) — compile-verified
//
#include <hip/hip_runtime.h>

// ---------------------------------------------------------------------------
// NeRF-style compositing for MI455X (gfx1250, wave32).
//   - 1 wave per ray, 16 rays per 512-thread block.
//   - sigma streamed global->LDS with GLOBAL_LOAD_ASYNC_TO_LDS_B128
//     (ASYNCcnt path), overlapped with the sort; rgb/inst prefetched
//     (global_prefetch_b8) so phase-C gathers hit cache.
//   - bitonic sort of (z,idx) packed u64 keys in LDS (block-cooperative).
//   - transmittance: exclusive cumprod done as exp(cumsum(log f)),
//     cumsum computed as a 16x128 @ 128x128 strictly-upper-triangular
//     ones-matrix product on V_WMMA_F32_16X16X4_F32 (fp32 tensor path).
//   - weighted reductions via wave32 shfl_xor trees.
// Memory-bound workload (~290 MB @ 23.3 TB/s => ~12.5 us floor).
// ---------------------------------------------------------------------------

#define S_SAMPLES   128
#define RAYS_PER_BLK 16
#define THREADS     (RAYS_PER_BLK * 32)
#define LSTRIDE     132            // padded f32 row stride (bank-conflict-free b64)
#define INF_DIST    1e10f
#define EPS_T       1e-10f

typedef float v2f __attribute__((ext_vector_type(2)));
typedef float v8f __attribute__((ext_vector_type(8)));

__global__ __launch_bounds__(THREADS)
void renderer_kernel(const float* __restrict__ z_vals,
                     const float* __restrict__ sigma_vals,
                     const float* __restrict__ rgb_vals,
                     const float* __restrict__ inst_vals,
                     const float* __restrict__ bg_color,
                     float* __restrict__ out_color,
                     float* __restrict__ out_depth,
                     float* __restrict__ out_inst,
                     float* __restrict__ out_w)
{
    __shared__ unsigned long long sortbuf[RAYS_PER_BLK][S_SAMPLES]; // 16 KB
    __shared__ float G [RAYS_PER_BLK][LSTRIDE];   // log(1 - alpha + eps)
    __shared__ float AL[RAYS_PER_BLK][LSTRIDE];   // alpha
    __shared__ float T [RAYS_PER_BLK][LSTRIDE];   // exclusive transmittance
    __shared__ __align__(16) float SG[RAYS_PER_BLK][S_SAMPLES];     // sigma stage

    const int tid  = threadIdx.x;
    const int lane = tid & 31;
    const int wav  = tid >> 5;                     // wave id == ray-in-block
    const int rayg = blockIdx.x * RAYS_PER_BLK + wav;

    const float* rgbR = rgb_vals  + (size_t)rayg * S_SAMPLES * 3;
    const float* insR = inst_vals + (size_t)rayg * S_SAMPLES * 3;

    // ---- async-stage sigma (512 B/ray) global -> LDS; completes during sort.
    // Each lane moves 16 B: global_load_async_to_lds_b128 (ASYNCcnt).
    {
        const float* gsrc = sigma_vals + (size_t)rayg * S_SAMPLES + lane * 4;
        unsigned ldsoff = (unsigned)(size_t)&SG[wav][lane * 4]; // low 32b of
                                                                // generic addr
                                                                // = LDS offset
        asm volatile("global_load_async_to_lds_b128 %0, %1, off"
                     :: "v"(ldsoff), "v"(gsrc) : "memory");
    }
    // ---- prefetch rgb/inst regions (12 x 128B lines each) for phase C ------
    if (lane < 12)      __builtin_prefetch(rgbR + lane * 32, 0, 1);
    else if (lane < 24) __builtin_prefetch(insR + (lane - 12) * 32, 0, 1);

    // ---- load depth keys (coalesced), pack (z_bits << 32) | sample_idx -----
    // z >= 0, so f32 bit pattern is order-preserving; idx gives stable ties.
    #pragma unroll
    for (int r = 0; r < 4; ++r) {
        const int s = lane + 32 * r;
        const float z = z_vals[(size_t)rayg * S_SAMPLES + s];
        sortbuf[wav][s] =
            ((unsigned long long)__float_as_uint(z) << 32) | (unsigned)s;
    }
    __syncthreads();

    // ---- bitonic sort of 128 keys per ray (ascending) ----------------------
    for (int k = 2; k <= S_SAMPLES; k <<= 1) {
        for (int j = k >> 1; j > 0; j >>= 1) {
            #pragma unroll
            for (int r = 0; r < 4; ++r) {
                const int i   = lane + 32 * r;
                const int ixj = i ^ j;
                if (ixj > i) {
                    const unsigned long long a = sortbuf[wav][i];
                    const unsigned long long b = sortbuf[wav][ixj];
                    const bool up = ((i & k) == 0);
                    if (up ? (a > b) : (a < b)) {
                        sortbuf[wav][i]   = b;
                        sortbuf[wav][ixj] = a;
                    }
                }
            }
            __syncthreads();
        }
    }

    // sigma staging is wave-local (this wave wrote SG[wav]); wait for it.
    asm volatile("s_wait_asynccnt 0x0" ::: "memory");

    // ---- per-sample: dists, tau, alpha, g = log(1 - alpha + eps) -----------
    #pragma unroll
    for (int r = 0; r < 4; ++r) {
        const int s = lane + 32 * r;
        const unsigned long long key = sortbuf[wav][s];
        const float z   = __uint_as_float((unsigned)(key >> 32));
        const int   idx = (int)(key & 0x7Full);
        float dist;
        if (s == S_SAMPLES - 1) {
            dist = INF_DIST;
        } else {
            const float zn =
                __uint_as_float((unsigned)(sortbuf[wav][s + 1] >> 32));
            dist = zn - z;
        }
        const float sg    = SG[wav][idx];
        const float tau   = fmaxf(sg, 0.0f) * dist;
        const float e     = expf(-tau);
        const float alpha = 1.0f - e;
        const float f     = e + EPS_T;            // 1 - alpha + eps
        G [wav][s] = logf(f);
        AL[wav][s] = alpha;
    }
    __syncthreads();

    // ---- exclusive cumsum of G along S via fp32 WMMA -----------------------
    // Y[ray][s] = sum_{t<s} G[ray][t]  ==  (G[16x128]) @ (U[128x128]),
    // U[t][s] = (t < s).  Tile j covers s in [16j, 16j+16); triangularity
    // means only k-blocks with k0 <= 16j+12 contribute.
    if (wav < 8) {                                // wave-uniform => EXEC all 1s
        const int jt   = wav;                     // output s-tile
        const int half = lane >> 4;               // 0: K pair {0,1}, 1: {2,3}
        const int m    = lane & 15;               // A row (ray) / B,D column
        const int koff = half * 2;
        const int scol = 16 * jt + m;             // global s for my B/D column
        const int kmax = 16 * jt + 12;

        v8f c = {0.f, 0.f, 0.f, 0.f, 0.f, 0.f, 0.f, 0.f};
        for (int k0 = 0; k0 <= kmax; k0 += 4) {
            const int t0 = k0 + koff;
            v2f a;                                // A 16x4 f32 fragment
            a.x = G[m][t0];
            a.y = G[m][t0 + 1];
            v2f b;                                // B 4x16: strict upper tri
            b.x = (t0     < scol) ? 1.0f : 0.0f;
            b.y = (t0 + 1 < scol) ? 1.0f : 0.0f;
#if __has_builtin(__builtin_amdgcn_wmma_f32_16x16x4_f32)
            c = __builtin_amdgcn_wmma_f32_16x16x4_f32(
                    false, a, false, b, (short)0, c, false, false);
#else
            // fallback (should not trigger on gfx1250 toolchains)
            #pragma unroll
            for (int v = 0; v < 8; ++v) { c[v] += a.x * b.x + a.y * b.y; }
#endif
        }
        // D 16x16 f32 layout: VGPR v, lanes 0-15 -> M=v, lanes 16-31 -> M=v+8
        #pragma unroll
        for (int v = 0; v < 8; ++v) {
            const int ray = v + half * 8;
            T[ray][16 * jt + m] = expf(c[v]);     // exclusive transmittance
        }
    }
    __syncthreads();

    // ---- weights, outputs, reductions --------------------------------------
    float dsum = 0.f;
    float c0 = 0.f, c1 = 0.f, c2 = 0.f;
    float i0 = 0.f, i1 = 0.f, i2 = 0.f;
    float nohit = 0.f;

    #pragma unroll
    for (int r = 0; r < 4; ++r) {
        const int s = lane + 32 * r;
        const unsigned long long key = sortbuf[wav][s];
        const float z   = __uint_as_float((unsigned)(key >> 32));
        const int   idx = (int)(key & 0x7Full);
        const float al  = AL[wav][s];
        const float tr  = T [wav][s];
        const float w   = al * tr;
        out_w[(size_t)rayg * S_SAMPLES + s] = w;   // coalesced per r
        dsum += w * z;
        c0 += w * rgbR[idx * 3 + 0];
        c1 += w * rgbR[idx * 3 + 1];
        c2 += w * rgbR[idx * 3 + 2];
        i0 += w * insR[idx * 3 + 0];
        i1 += w * insR[idx * 3 + 1];
        i2 += w * insR[idx * 3 + 2];
        if (s == S_SAMPLES - 1) nohit = tr * (1.0f - al + EPS_T); // trans[-1]
    }

    // wave32 reductions
    #pragma unroll
    for (int off = 16; off > 0; off >>= 1) {
        dsum  += __shfl_xor(dsum,  off, 32);
        c0    += __shfl_xor(c0,    off, 32);
        c1    += __shfl_xor(c1,    off, 32);
        c2    += __shfl_xor(c2,    off, 32);
        i0    += __shfl_xor(i0,    off, 32);
        i1    += __shfl_xor(i1,    off, 32);
        i2    += __shfl_xor(i2,    off, 32);
        nohit += __shfl_xor(nohit, off, 32);   // single non-zero contributor
    }

    if (lane == 0) {
        out_depth[rayg] = dsum;
        out_color[(size_t)rayg * 3 + 0] = c0 + nohit * bg_color[(size_t)rayg * 3 + 0];
        out_color[(size_t)rayg * 3 + 1] = c1 + nohit * bg_color[(size_t)rayg * 3 + 1];
        out_color[(size_t)rayg * 3 + 2] = c2 + nohit * bg_color[(size_t)rayg * 3 + 2];
        out_inst [(size_t)rayg * 3 + 0] = i0;
        out_inst [(size_t)rayg * 3 + 1] = i1;
        out_inst [(size_t)rayg * 3 + 2] = i2;
    }
}

extern "C" void kernel_launch(void* const* d_in, const int* in_sizes, int n_in,
                              void* d_out, int out_size, void* d_ws, size_t ws_size,
                              hipStream_t stream)
{
    (void)n_in; (void)out_size; (void)d_ws; (void)ws_size;
    const float* z_vals     = (const float*)d_in[0];
    const float* sigma_vals = (const float*)d_in[1];
    const float* rgb_vals   = (const float*)d_in[2];
    const float* inst_vals  = (const float*)d_in[3];
    const float* bg_color   = (const float*)d_in[4];

    const int N = in_sizes[0] / S_SAMPLES;        // 65536

    // output layout: color [N,3] | depth [N] | instance [N,3] | w [N,128]
    float* out       = (float*)d_out;
    float* out_color = out;
    float* out_depth = out_color + (size_t)N * 3;
    float* out_inst  = out_depth + (size_t)N;
    float* out_w     = out_inst  + (size_t)N * 3;

    dim3 grid(N / RAYS_PER_BLK);
    dim3 block(THREADS);
    hipLaunchKernelGGL(renderer_kernel, grid, block, 0, stream,
                       z_vals, sigma_vals, rgb_vals, inst_vals, bg_color,
                       out_color, out_depth, out_inst, out_w);
}